// HungarianMatchingLoss_43499428774040
// MI455X (gfx1250) — compile-verified
//
#include <hip/hip_runtime.h>
#include <hip/hip_bf16.h>
#include <math.h>

// Problem constants (fixed by the reference)
#define B_   64
#define Q_   300
#define QP   304      // Q padded to multiple of 16 for WMMA tiles
#define N_   64
#define C_   6
#define NT   384      // 12 wave32s per block; >= Q_ so one column per thread
#define EPSF 1e-6f
#define INF_ 3.0e38f

typedef float v2f __attribute__((ext_vector_type(2)));
typedef float v8f __attribute__((ext_vector_type(8)));

__device__ __forceinline__ v8f wmma_f32_16x16x4(v2f a, v2f b, v8f c) {
  // D = A(16x4 f32) * B(4x16 f32) + C(16x16 f32); exact for one-hot A.
  return __builtin_amdgcn_wmma_f32_16x16x4_f32(
      /*neg_a=*/false, a, /*neg_b=*/false, b,
      /*c_mod=*/(short)0, c, /*reuse_a=*/false, /*reuse_b=*/false);
}

__global__ __launch_bounds__(NT, 1)
void hml_batch_kernel(const float* __restrict__ exists,
                      const float* __restrict__ coords,
                      const float* __restrict__ width,
                      const float* __restrict__ ef_logits,
                      const float* __restrict__ tracks,
                      const unsigned char* __restrict__ track_mask,
                      float* __restrict__ accf, int* __restrict__ acci) {
  // ---- LDS-resident state (~102 KB; WGP has 320 KB) ----
  __shared__ float sCost[N_][QP];      // assignment cost, rows = GT, cols = preds
  __shared__ float sNlpT[8][QP];       // -log_softmax, class-major; rows 6,7 + cols >=300 zero
  __shared__ float sCo[QP][4];
  __shared__ float sW[QP];
  __shared__ float sCE[QP];            // -log(clip(pe)+1e-8)   (cost term)
  __shared__ float sNLP[QP];           // -log(clip(pe))        (exists loss)
  __shared__ float sNL1[QP];           // -log(1-clip(pe))      (noobj loss)
  __shared__ float sGt[N_][5];
  __shared__ int   sCls[N_];
  // JV matching state
  __shared__ float sShort[QP];
  __shared__ float sV[QP];
  __shared__ int   sPath[QP];
  __shared__ int   sR4C[QP];
  __shared__ unsigned char sSC[QP];
  __shared__ float sU[N_];
  __shared__ int   sC4R[N_];
  __shared__ int   sSR[N_];
  __shared__ float sRedV[NT / 32];
  __shared__ int   sRedI[NT / 32];
  __shared__ int   sBI, sSink, sNV;
  __shared__ float sMinVal;

  const int b   = blockIdx.x;
  const int tid = threadIdx.x;

  // ---------------- Phase 1: load + per-prediction precompute ----------------
  for (int q = tid; q < Q_; q += NT) {
    float pe = exists[(size_t)b * Q_ + q];
    pe = fminf(fmaxf(pe, EPSF), 1.0f - EPSF);
    sCE[q]  = -logf(pe + 1e-8f);
    sNLP[q] = -logf(pe);
    sNL1[q] = -logf(1.0f - pe);
    const float* cp = coords + ((size_t)b * Q_ + q) * 4;
    sCo[q][0] = cp[0]; sCo[q][1] = cp[1]; sCo[q][2] = cp[2]; sCo[q][3] = cp[3];
    sW[q] = width[(size_t)b * Q_ + q];
    const float* lp = ef_logits + ((size_t)b * Q_ + q) * C_;
    float l0 = lp[0], l1 = lp[1], l2 = lp[2], l3 = lp[3], l4 = lp[4], l5 = lp[5];
    float mx = fmaxf(fmaxf(fmaxf(l0, l1), fmaxf(l2, l3)), fmaxf(l4, l5));
    float s = expf(l0 - mx) + expf(l1 - mx) + expf(l2 - mx) +
              expf(l3 - mx) + expf(l4 - mx) + expf(l5 - mx);
    float lse = mx + logf(s);
    sNlpT[0][q] = lse - l0; sNlpT[1][q] = lse - l1; sNlpT[2][q] = lse - l2;
    sNlpT[3][q] = lse - l3; sNlpT[4][q] = lse - l4; sNlpT[5][q] = lse - l5;
  }
  // Zero-pad the K dimension (classes 6,7) and q in [300,304) for WMMA.
  for (int x = tid; x < 8 * QP; x += NT) {
    int c = x / QP, q = x - c * QP;
    if (c >= C_ || q >= Q_) sNlpT[c][q] = 0.0f;
  }
  for (int n = tid; n < N_; n += NT) {
    const float* tp = tracks + ((size_t)b * N_ + n) * 6;
    sGt[n][0] = tp[0]; sGt[n][1] = tp[1]; sGt[n][2] = tp[2];
    sGt[n][3] = tp[3]; sGt[n][4] = tp[4];
    sCls[n] = (int)tp[5];
  }
  if (tid == 0) {
    int nv = 0;
    for (int n = 0; n < N_; ++n) nv += track_mask[(size_t)b * N_ + n] ? 1 : 0;
    sNV = nv;
  }
  __syncthreads();

  // ---------------- Phase 2: base cost (coord + width + exists) ----------------
  for (int idx = tid; idx < N_ * Q_; idx += NT) {
    int n = idx / Q_, q = idx - n * Q_;
    float cc = fabsf(sCo[q][0] - sGt[n][0]) + fabsf(sCo[q][1] - sGt[n][1]) +
               fabsf(sCo[q][2] - sGt[n][2]) + fabsf(sCo[q][3] - sGt[n][3]);
    float cw = fabsf(sW[q] - sGt[n][4]);
    sCost[n][q] = 5.0f * cc + 2.0f * cw + 2.0f * sCE[q];
  }
  __syncthreads();

  // ---------------- Phase 3: class-cost via one-hot WMMA GEMM ----------------
  // cost_ef[n,q] = sum_c onehot[n,c] * (-logp[q,c]) -> V_WMMA_F32_16X16X4_F32
  {
    const int wid  = tid >> 5;
    const int lane = tid & 31;
    const int lm   = lane & 15;   // row of A / col of B within tile
    const int lh   = lane >> 4;   // half selector: K pair / M offset
    for (int tile = wid; tile < 4 * 19; tile += NT / 32) {   // wave-uniform loop
      const int mt = tile / 19, nt = tile - mt * 19;
      const int m  = mt * 16 + lm;
      const int q0 = nt * 16;
      const int cls = sCls[m];
      v8f acc = {0.f, 0.f, 0.f, 0.f, 0.f, 0.f, 0.f, 0.f};
      // K-step 0 covers classes 0..3 (lanes 0-15: K=0,1; lanes 16-31: K=2,3)
      v2f a0, b0;
      a0.x = (cls == lh * 2 + 0) ? 1.0f : 0.0f;
      a0.y = (cls == lh * 2 + 1) ? 1.0f : 0.0f;
      b0.x = sNlpT[lh * 2 + 0][q0 + lm];
      b0.y = sNlpT[lh * 2 + 1][q0 + lm];
      acc = wmma_f32_16x16x4(a0, b0, acc);
      // K-step 1 covers classes 4..7 (6,7 are zero padding)
      v2f a1, b1;
      a1.x = (cls == 4 + lh * 2) ? 1.0f : 0.0f;
      a1.y = (cls == 5 + lh * 2) ? 1.0f : 0.0f;
      b1.x = sNlpT[4 + lh * 2][q0 + lm];
      b1.y = sNlpT[5 + lh * 2][q0 + lm];
      acc = wmma_f32_16x16x4(a1, b1, acc);
      // D layout: VGPR r, lane l -> M = r + 8*(l/16), Ncol = l%16
#pragma unroll
      for (int r = 0; r < 8; ++r) {
        sCost[mt * 16 + r + 8 * lh][q0 + lm] += 2.0f * acc[r];  // L_EF = 2
      }
    }
  }
  __syncthreads();

  // ---------------- Phase 4: JV shortest-augmenting-path assignment ----------------
  for (int j = tid; j < Q_; j += NT) { sV[j] = 0.0f; sR4C[j] = -1; }
  for (int r = tid; r < N_; r += NT) { sU[r] = 0.0f; sC4R[r] = -1; }
  __syncthreads();

  const int nv = sNV;
  for (int row = 0; row < nv; ++row) {
    for (int j = tid; j < Q_; j += NT) { sShort[j] = INF_; sSC[j] = 0; sPath[j] = -1; }
    for (int r = tid; r < N_; r += NT) sSR[r] = 0;
    if (tid == 0) { sBI = row; sMinVal = 0.0f; sSink = -1; }
    __syncthreads();

    while (true) {
      const int   i    = sBI;
      const float minv = sMinVal;
      if (tid == 0) sSR[i] = 1;
      // Relax all remaining columns in parallel (one column per thread).
      if (tid < Q_ && !sSC[tid]) {
        float d = minv + sCost[i][tid] - sU[i] - sV[tid];
        if (d < sShort[tid]) { sShort[tid] = d; sPath[tid] = i; }
      }
      __syncthreads();
      // argmin over remaining columns (first-index tie-break, as numpy).
      float val = (tid < Q_ && !sSC[tid]) ? sShort[tid] : INF_;
      int   idx = tid;
      for (int off = 16; off > 0; off >>= 1) {
        float ov = __shfl_down(val, off);
        int   oi = __shfl_down(idx, off);
        if (ov < val || (ov == val && oi < idx)) { val = ov; idx = oi; }
      }
      if ((tid & 31) == 0) { sRedV[tid >> 5] = val; sRedI[tid >> 5] = idx; }
      __syncthreads();
      if (tid == 0) {
        float bv = sRedV[0]; int bj = sRedI[0];
        for (int w = 1; w < NT / 32; ++w)
          if (sRedV[w] < bv) { bv = sRedV[w]; bj = sRedI[w]; }
        sMinVal = bv;
        sSC[bj] = 1;
        if (sR4C[bj] == -1) sSink = bj; else sBI = sR4C[bj];
      }
      __syncthreads();
      if (sSink != -1) break;
    }

    // Dual updates.
    const float minv = sMinVal;
    if (tid == 0) sU[row] += minv;
    if (tid < N_ && tid != row && sSR[tid]) sU[tid] += minv - sShort[sC4R[tid]];
    if (tid < Q_ && sSC[tid]) sV[tid] -= minv - sShort[tid];
    __syncthreads();
    // Augment along the alternating path (serial, thread 0).
    if (tid == 0) {
      int j = sSink;
      while (true) {
        int i = sPath[j];
        sR4C[j] = i;
        int t = sC4R[i];
        sC4R[i] = j;
        j = t;
        if (i == row) break;
      }
    }
    __syncthreads();
  }

  // ---------------- Phase 5: loss reduction for this batch ----------------
  float coordS = 0.f, widthS = 0.f, efS = 0.f, exS = 0.f, noobjS = 0.f;
  for (int n = tid; n < nv; n += NT) {
    int p = sC4R[n];
    coordS += fabsf(sCo[p][0] - sGt[n][0]) + fabsf(sCo[p][1] - sGt[n][1]) +
              fabsf(sCo[p][2] - sGt[n][2]) + fabsf(sCo[p][3] - sGt[n][3]);
    widthS += fabsf(sW[p] - sGt[n][4]);
    efS    += sNlpT[sCls[n]][p];   // = NLL of matched class
    exS    += sNLP[p];
  }
  for (int q = tid; q < Q_; q += NT) {
    if (sR4C[q] < 0) noobjS += sNL1[q];
  }
  // Wave32 sum-reduce, then one atomic per wave per term.
  for (int off = 16; off > 0; off >>= 1) {
    coordS += __shfl_down(coordS, off);
    widthS += __shfl_down(widthS, off);
    efS    += __shfl_down(efS, off);
    exS    += __shfl_down(exS, off);
    noobjS += __shfl_down(noobjS, off);
  }
  if ((tid & 31) == 0) {
    atomicAdd(&accf[0], coordS);
    atomicAdd(&accf[1], widthS);
    atomicAdd(&accf[2], efS);
    atomicAdd(&accf[3], exS);
    atomicAdd(&accf[4], noobjS);
  }
  if (tid == 0) atomicAdd(&acci[5], nv);
}

__global__ void hml_init_kernel(float* accf) {
  if (threadIdx.x < 8) accf[threadIdx.x] = 0.0f;  // zero bits == int 0 too
}

__global__ void hml_final_kernel(const float* __restrict__ accf,
                                 const int* __restrict__ acci,
                                 float* __restrict__ out, int BQ) {
  if (threadIdx.x == 0 && blockIdx.x == 0) {
    int nm_i = acci[5];
    float nm = (float)((nm_i > 1) ? nm_i : 1);
    int nu_i = BQ - nm_i;
    float nu = (float)((nu_i > 1) ? nu_i : 1);
    float coord = 5.0f * accf[0] / nm;
    float wid   = 2.0f * accf[1] / nm;
    float ef    = 2.0f * accf[2] / nm;
    float ex    = 2.0f * accf[3] / nm;
    float no    = 1.0f * accf[4] / nu;
    out[0] = coord + wid + ef + ex + no;
    out[1] = coord; out[2] = wid; out[3] = ef; out[4] = ex; out[5] = no;
  }
}

extern "C" void kernel_launch(void* const* d_in, const int* in_sizes, int n_in,
                              void* d_out, int out_size, void* d_ws, size_t ws_size,
                              hipStream_t stream) {
  const float*         exists = (const float*)d_in[0];         // (B,Q,1) f32
  const float*         coords = (const float*)d_in[1];         // (B,Q,4) f32
  const float*         width  = (const float*)d_in[2];         // (B,Q,1) f32
  const float*         eflog  = (const float*)d_in[3];         // (B,Q,6) f32
  const float*         tracks = (const float*)d_in[4];         // (B,N,6) f32
  const unsigned char* mask   = (const unsigned char*)d_in[5]; // (B,N) bool (1B)
  float* accf = (float*)d_ws;
  int*   acci = (int*)d_ws;
  float* out  = (float*)d_out;

  hml_init_kernel<<<1, 32, 0, stream>>>(accf);
  hml_batch_kernel<<<B_, NT, 0, stream>>>(exists, coords, width, eflog,
                                          tracks, mask, accf, acci);
  hml_final_kernel<<<1, 32, 0, stream>>>(accf, acci, out, B_ * Q_);
}